// GeometricMHA_87084756894162
// MI455X (gfx1250) — compile-verified
//
#include <hip/hip_runtime.h>
#include <math.h>

#define B_ 2
#define L_ 1024
#define E_ 1024
#define H_ 16
#define QKV_N 240   // H*15
#define OUT_K 48    // H*3

typedef __attribute__((ext_vector_type(2))) float v2f;
typedef __attribute__((ext_vector_type(8))) float v8f;

__device__ __forceinline__ v8f wmma4(v2f a, v2f b, v8f c) {
  // D = A(16x4,f32) x B(4x16,f32) + C(16x16,f32)
  return __builtin_amdgcn_wmma_f32_16x16x4_f32(false, a, false, b, (short)0, c, false, false);
}

// ---------------- 0) LDS-tiled transpose: in(rows x cols) -> out(cols x rows) ----------
__global__ void transpose_kernel(const float* __restrict__ in, float* __restrict__ out,
                                 int rows, int cols) {
  __shared__ float tile[32][33];
  const int c0 = blockIdx.x * 32, r0 = blockIdx.y * 32;
  const int tx = threadIdx.x & 31, ty = threadIdx.x >> 5;   // 256 thr = 32x8
  for (int i = ty; i < 32; i += 8) {
    const int r = r0 + i, c = c0 + tx;
    tile[i][tx] = (r < rows && c < cols) ? in[(size_t)r * cols + c] : 0.f;
  }
  __syncthreads();
  for (int i = ty; i < 32; i += 8) {
    const int r = c0 + i, c = r0 + tx;                      // out is cols x rows
    if (r < cols && c < rows) out[(size_t)r * rows + c] = tile[tx][i];
  }
}

// ---------------- 1) LayerNorm ----------------
__global__ void ln_kernel(const float* __restrict__ x, const float* __restrict__ gamma,
                          const float* __restrict__ beta, float* __restrict__ xn) {
  __shared__ float red[256];
  const int row = blockIdx.x;                 // 0 .. B*L-1
  const float* xr = x + (size_t)row * E_;
  float s = 0.f;
  for (int i = threadIdx.x; i < E_; i += 256) s += xr[i];
  red[threadIdx.x] = s; __syncthreads();
  for (int o = 128; o > 0; o >>= 1) { if (threadIdx.x < o) red[threadIdx.x] += red[threadIdx.x + o]; __syncthreads(); }
  const float mu = red[0] * (1.0f / E_);
  __syncthreads();
  float s2 = 0.f;
  for (int i = threadIdx.x; i < E_; i += 256) { float d = xr[i] - mu; s2 += d * d; }
  red[threadIdx.x] = s2; __syncthreads();
  for (int o = 128; o > 0; o >>= 1) { if (threadIdx.x < o) red[threadIdx.x] += red[threadIdx.x + o]; __syncthreads(); }
  const float inv = rsqrtf(red[0] * (1.0f / E_) + 1e-5f);
  float* xo = xn + (size_t)row * E_;
  for (int i = threadIdx.x; i < E_; i += 256)
    xo[i] = (xr[i] - mu) * inv * gamma[i] + beta[i];
}

// ---------------- 2) QKV GEMM: (2048 x 1024) @ (1024 x 240), Wt is (240 x 1024) ------
// One wave computes a 16(M) x 48(N) strip: one A-frag feeds 3 WMMAs.
__global__ void gemm_qkv(const float* __restrict__ A, const float* __restrict__ Wt,
                         float* __restrict__ C) {
  const int K = E_, N = QKV_N;
  const int lane = threadIdx.x, lo = lane & 15, kh = lane >> 4;
  const int m0 = blockIdx.x * 16, n0 = blockIdx.y * 48;
  const float* arow = A + (size_t)(m0 + lo) * K + 2 * kh;
  const float* b0 = Wt + (size_t)(n0 + lo) * K + 2 * kh;        // column n0+lo
  const float* b1 = Wt + (size_t)(n0 + 16 + lo) * K + 2 * kh;
  const float* b2 = Wt + (size_t)(n0 + 32 + lo) * K + 2 * kh;
  v8f acc0 = {}, acc1 = {}, acc2 = {};
  for (int k = 0; k < K; k += 4) {
    __builtin_prefetch(arow + k + 128, 0, 1);   // global_prefetch_b8
    __builtin_prefetch(b0 + k + 128, 0, 1);
    const v2f a  = *(const v2f*)(arow + k);
    const v2f w0 = *(const v2f*)(b0 + k);
    const v2f w1 = *(const v2f*)(b1 + k);
    const v2f w2 = *(const v2f*)(b2 + k);
    acc0 = wmma4(a, w0, acc0);
    acc1 = wmma4(a, w1, acc1);
    acc2 = wmma4(a, w2, acc2);
  }
#pragma unroll
  for (int r = 0; r < 8; ++r) {
    const size_t row = (size_t)(m0 + r + 8 * kh) * N;
    C[row + n0 + lo]      = acc0[r];
    C[row + n0 + 16 + lo] = acc1[r];
    C[row + n0 + 32 + lo] = acc2[r];
  }
}

// ---------------- 3) rotate into global frame, +translations, pad to 4 ----------------
__global__ void rot_fwd(const float* __restrict__ qkvraw, const float* __restrict__ rot,
                        const float* __restrict__ trans,
                        float* __restrict__ Qr, float* __restrict__ Kr,
                        float* __restrict__ Qd, float* __restrict__ Kd,
                        float* __restrict__ Vv,
                        float* __restrict__ qn2, float* __restrict__ kn2) {
  const int idx = blockIdx.x * blockDim.x + threadIdx.x;   // (bl, f, h)
  const int total = B_ * L_ * 5 * H_;
  if (idx >= total) return;
  const int h = idx % H_;
  const int t = idx / H_;
  const int f = t % 5;
  const int bl = t / 5;
  const float* v = qkvraw + (size_t)bl * QKV_N + (size_t)(f * H_ + h) * 3;
  const float* R = rot + (size_t)bl * 9;
  const float x = v[0], y = v[1], z = v[2];
  float o0 = R[0] * x + R[1] * y + R[2] * z;
  float o1 = R[3] * x + R[4] * y + R[5] * z;
  float o2 = R[6] * x + R[7] * y + R[8] * z;
  if (f == 2 || f == 3) {
    const float* T = trans + (size_t)bl * 3;
    o0 += T[0]; o1 += T[1]; o2 += T[2];
  }
  const int b = bl / L_, l = bl % L_;
  const size_t di = ((size_t)(b * H_ + h) * L_ + l) * 4;
  float* dst = (f == 0) ? Qr : (f == 1) ? Kr : (f == 2) ? Qd : (f == 3) ? Kd : Vv;
  dst[di + 0] = o0; dst[di + 1] = o1; dst[di + 2] = o2; dst[di + 3] = 0.f;
  if (f == 2) qn2[(size_t)(b * H_ + h) * L_ + l] = o0 * o0 + o1 * o1 + o2 * o2;
  if (f == 3) kn2[(size_t)(b * H_ + h) * L_ + l] = o0 * o0 + o1 * o1 + o2 * o2;
}

// ---------------- 4) fused flash attention, one wave per (b,h,i-tile) ----------------
__global__ void attn_kernel(const float* __restrict__ Qr, const float* __restrict__ Kr,
                            const float* __restrict__ Qd, const float* __restrict__ Kd,
                            const float* __restrict__ Vv,
                            const float* __restrict__ qn2, const float* __restrict__ kn2,
                            const unsigned char* __restrict__ mask,
                            const float* __restrict__ r_scale, const float* __restrict__ d_scale,
                            float* __restrict__ out /* (B,L,H,3) */) {
  const int it = blockIdx.x, h = blockIdx.y, b = blockIdx.z;
  const int lane = threadIdx.x, lo = lane & 15, kh = lane >> 4;
  const size_t base = (size_t)(b * H_ + h) * L_;
  const float* qrp = Qr + base * 4;
  const float* qdp = Qd + base * 4;
  const float* krp = Kr + base * 4;
  const float* kdp = Kd + base * 4;
  const float* vp  = Vv + base * 4;
  const float scale = 0.57735026918962576f;  // 3^-0.5
  const float rs = r_scale[h], ds = d_scale[h];
  const float cr = ((rs > 20.f) ? rs : log1pf(__expf(rs))) * scale;  // softplus
  const float cd = ((ds > 20.f) ? ds : log1pf(__expf(ds))) * scale;

  const int i0 = it * 16;
  const v2f aqr = *(const v2f*)(qrp + (size_t)(i0 + lo) * 4 + 2 * kh);
  const v2f aqd = *(const v2f*)(qdp + (size_t)(i0 + lo) * 4 + 2 * kh);

  float qn[8], mbi[8];
#pragma unroll
  for (int r = 0; r < 8; ++r) {
    const int m = i0 + r + 8 * kh;
    qn[r]  = qn2[base + m];
    mbi[r] = mask[(size_t)b * L_ + m] ? -1e9f : 0.f;
  }
  float Mx[8], Sm[8], O0[8], O1[8], O2[8];
#pragma unroll
  for (int r = 0; r < 8; ++r) { Mx[r] = -3.0e38f; Sm[r] = 0.f; O0[r] = 0.f; O1[r] = 0.f; O2[r] = 0.f; }

  for (int jt = 0; jt < L_ / 16; ++jt) {
    const int j = jt * 16 + lo;
    const v2f bkr = *(const v2f*)(krp + (size_t)j * 4 + 2 * kh);
    const v2f bkd = *(const v2f*)(kdp + (size_t)j * 4 + 2 * kh);
    v8f z = {};
    const v8f sr = wmma4(aqr, bkr, z);   // Q_r . K_r per (m,n)
    const v8f sd = wmma4(aqd, bkd, z);   // Q_d . K_d per (m,n)
    const float knj = kn2[base + j];
    const float mbj = mask[(size_t)b * L_ + j] ? -1e9f : 0.f;
    const float v0 = vp[(size_t)j * 4 + 0];
    const float v1 = vp[(size_t)j * 4 + 1];
    const float v2 = vp[(size_t)j * 4 + 2];
#pragma unroll
    for (int r = 0; r < 8; ++r) {
      const float d2 = qn[r] + knj - 2.f * sd[r];
      const float dist = sqrtf(fmaxf(d2, 0.f));
      const float logit = sr[r] * cr - dist * cd + mbi[r] + mbj;
      const float mnew = fmaxf(Mx[r], logit);
      const float corr = __expf(Mx[r] - mnew);
      const float p = __expf(logit - mnew);
      Sm[r] = Sm[r] * corr + p;
      O0[r] = O0[r] * corr + p * v0;
      O1[r] = O1[r] * corr + p * v1;
      O2[r] = O2[r] * corr + p * v2;
      Mx[r] = mnew;
    }
  }

  // flash-combine across the 16 lanes that share each output row (wave32 xor shuffles)
#pragma unroll
  for (int s = 1; s <= 8; s <<= 1) {
#pragma unroll
    for (int r = 0; r < 8; ++r) {
      const float mo = __shfl_xor(Mx[r], s, 32);
      const float so = __shfl_xor(Sm[r], s, 32);
      const float a0 = __shfl_xor(O0[r], s, 32);
      const float a1 = __shfl_xor(O1[r], s, 32);
      const float a2 = __shfl_xor(O2[r], s, 32);
      const float mnew = fmaxf(Mx[r], mo);
      const float c1 = __expf(Mx[r] - mnew);
      const float c2 = __expf(mo - mnew);
      Sm[r] = Sm[r] * c1 + so * c2;
      O0[r] = O0[r] * c1 + a0 * c2;
      O1[r] = O1[r] * c1 + a1 * c2;
      O2[r] = O2[r] * c1 + a2 * c2;
      Mx[r] = mnew;
    }
  }

  if (lo == 0) {
#pragma unroll
    for (int r = 0; r < 8; ++r) {
      const int m = i0 + r + 8 * kh;
      const float inv = 1.f / Sm[r];
      const size_t o = (((size_t)b * L_ + m) * H_ + h) * 3;
      out[o + 0] = O0[r] * inv;
      out[o + 1] = O1[r] * inv;
      out[o + 2] = O2[r] * inv;
    }
  }
}

// ---------------- 5) rotate back with R^T, merge heads ----------------
__global__ void rot_back(const float* __restrict__ attn /* (B,L,H,3) */,
                         const float* __restrict__ rot,
                         float* __restrict__ merged /* (B*L, 48) */) {
  const int idx = blockIdx.x * blockDim.x + threadIdx.x;
  const int total = B_ * L_ * H_;
  if (idx >= total) return;
  const int h = idx % H_;
  const int bl = idx / H_;
  const float* o = attn + ((size_t)bl * H_ + h) * 3;
  const float* R = rot + (size_t)bl * 9;
  const float x = o[0], y = o[1], z = o[2];
  merged[(size_t)bl * OUT_K + h * 3 + 0] = R[0] * x + R[3] * y + R[6] * z;
  merged[(size_t)bl * OUT_K + h * 3 + 1] = R[1] * x + R[4] * y + R[7] * z;
  merged[(size_t)bl * OUT_K + h * 3 + 2] = R[2] * x + R[5] * y + R[8] * z;
}

// ---------------- 6) output GEMM: (2048 x 48) @ (48 x 1024) + bias -------------------
// Wt is Wout^T (1024 x 48). One wave computes a 16(M) x 64(N) strip; K fully unrolled.
__global__ void gemm_out(const float* __restrict__ A, const float* __restrict__ Wt,
                         const float* __restrict__ bias, float* __restrict__ C) {
  const int lane = threadIdx.x, lo = lane & 15, kh = lane >> 4;
  const int m0 = blockIdx.x * 16, n0 = blockIdx.y * 64;
  const float* arow = A + (size_t)(m0 + lo) * OUT_K + 2 * kh;
  v8f acc0 = {}, acc1 = {}, acc2 = {}, acc3 = {};
#pragma unroll
  for (int k = 0; k < OUT_K; k += 4) {
    const v2f a = *(const v2f*)(arow + k);
    const v2f w0 = *(const v2f*)(Wt + (size_t)(n0 + lo) * OUT_K + k + 2 * kh);
    const v2f w1 = *(const v2f*)(Wt + (size_t)(n0 + 16 + lo) * OUT_K + k + 2 * kh);
    const v2f w2 = *(const v2f*)(Wt + (size_t)(n0 + 32 + lo) * OUT_K + k + 2 * kh);
    const v2f w3 = *(const v2f*)(Wt + (size_t)(n0 + 48 + lo) * OUT_K + k + 2 * kh);
    acc0 = wmma4(a, w0, acc0);
    acc1 = wmma4(a, w1, acc1);
    acc2 = wmma4(a, w2, acc2);
    acc3 = wmma4(a, w3, acc3);
  }
#pragma unroll
  for (int r = 0; r < 8; ++r) {
    const size_t row = (size_t)(m0 + r + 8 * kh) * E_;
    C[row + n0 + lo]      = acc0[r] + bias[n0 + lo];
    C[row + n0 + 16 + lo] = acc1[r] + bias[n0 + 16 + lo];
    C[row + n0 + 32 + lo] = acc2[r] + bias[n0 + 32 + lo];
    C[row + n0 + 48 + lo] = acc3[r] + bias[n0 + 48 + lo];
  }
}

extern "C" void kernel_launch(void* const* d_in, const int* in_sizes, int n_in,
                              void* d_out, int out_size, void* d_ws, size_t ws_size,
                              hipStream_t stream) {
  (void)in_sizes; (void)n_in; (void)out_size; (void)ws_size;
  const float* x       = (const float*)d_in[0];
  const float* rot     = (const float*)d_in[1];
  const float* trans   = (const float*)d_in[2];
  const unsigned char* mask = (const unsigned char*)d_in[3];
  const float* Wqkv    = (const float*)d_in[4];
  const float* Wout    = (const float*)d_in[5];
  const float* bout    = (const float*)d_in[6];
  const float* gamma   = (const float*)d_in[7];
  const float* beta    = (const float*)d_in[8];
  const float* r_scale = (const float*)d_in[9];
  const float* d_scale = (const float*)d_in[10];
  float* out = (float*)d_out;

  float* ws = (float*)d_ws;
  size_t off = 0;
  float* xn     = ws + off; off += (size_t)B_ * L_ * E_;        // 2M floats
  float* qkvraw = ws + off; off += (size_t)B_ * L_ * QKV_N;
  float* Qr     = ws + off; off += (size_t)B_ * H_ * L_ * 4;
  float* Kr     = ws + off; off += (size_t)B_ * H_ * L_ * 4;
  float* Qd     = ws + off; off += (size_t)B_ * H_ * L_ * 4;
  float* Kd     = ws + off; off += (size_t)B_ * H_ * L_ * 4;
  float* Vv     = ws + off; off += (size_t)B_ * H_ * L_ * 4;
  float* qn2    = ws + off; off += (size_t)B_ * H_ * L_;
  float* kn2    = ws + off; off += (size_t)B_ * H_ * L_;
  float* attn   = ws + off; off += (size_t)B_ * L_ * H_ * 3;
  float* merged = ws + off; off += (size_t)B_ * L_ * OUT_K;
  float* WqkvT  = ws + off; off += (size_t)QKV_N * E_;          // 240 x 1024
  float* WoutT  = ws + off; off += (size_t)E_ * OUT_K;          // 1024 x 48

  // weight transposes (coalesced B-frag loads in the WMMA GEMMs)
  transpose_kernel<<<dim3((QKV_N + 31) / 32, (E_ + 31) / 32), 256, 0, stream>>>(Wqkv, WqkvT, E_, QKV_N);
  transpose_kernel<<<dim3((E_ + 31) / 32, (OUT_K + 31) / 32), 256, 0, stream>>>(Wout, WoutT, OUT_K, E_);

  ln_kernel<<<B_ * L_, 256, 0, stream>>>(x, gamma, beta, xn);
  gemm_qkv<<<dim3((B_ * L_) / 16, QKV_N / 48), 32, 0, stream>>>(xn, WqkvT, qkvraw);
  {
    const int tot = B_ * L_ * 5 * H_;
    rot_fwd<<<(tot + 255) / 256, 256, 0, stream>>>(qkvraw, rot, trans, Qr, Kr, Qd, Kd, Vv, qn2, kn2);
  }
  attn_kernel<<<dim3(L_ / 16, H_, B_), 32, 0, stream>>>(Qr, Kr, Qd, Kd, Vv, qn2, kn2,
                                                        mask, r_scale, d_scale, attn);
  {
    const int tot = B_ * L_ * H_;
    rot_back<<<(tot + 255) / 256, 256, 0, stream>>>(attn, rot, merged);
  }
  gemm_out<<<dim3((B_ * L_) / 16, E_ / 64), 32, 0, stream>>>(merged, WoutT, bout, out);
}